// yolov1_loss_36103495090632
// MI455X (gfx1250) — compile-verified
//
#include <hip/hip_runtime.h>

// ---------------------------------------------------------------------------
// YOLOv1 loss on MI455X (gfx1250): pure streaming reduction (193 MB -> 5 f32).
// Memory-bound: ~8.3 us floor at 23.3 TB/s. No matmul -> WMMA inapplicable.
// CDNA5 path used: async global->LDS (ASYNCcnt) double-buffered tile pipeline.
// ---------------------------------------------------------------------------

#define CH          30
#define TILE        128                    // cells per tile
#define NTHR        64                     // 2 wave32 per block
#define TILE_FLOATS (TILE * CH)            // 3840
#define TILE_BYTES  (TILE_FLOATS * 4)      // 15360
// b128 chunks per array per tile = 960; 960 / 64 threads = 15 exactly
// -> every wave issues exactly 15 async ops per array, 30 per tile.
#define PER_TILE_ASYNC 30

#if defined(__has_builtin)
#  if __has_builtin(__builtin_amdgcn_global_load_async_to_lds_b128)
#    define HAVE_ASYNC_LDS 1
#  endif
#  if __has_builtin(__builtin_amdgcn_s_wait_asynccnt)
#    define HAVE_WAIT_BUILTIN 1
#  endif
#endif
#ifndef HAVE_ASYNC_LDS
#  define HAVE_ASYNC_LDS 0
#endif
#ifndef HAVE_WAIT_BUILTIN
#  define HAVE_WAIT_BUILTIN 0
#endif

#define AS1 __attribute__((address_space(1)))
#define AS3 __attribute__((address_space(3)))

typedef int v4i __attribute__((ext_vector_type(4)));

__device__ __forceinline__ void async_cp16(void* lds_dst, const void* gsrc) {
#if HAVE_ASYNC_LDS
    __builtin_amdgcn_global_load_async_to_lds_b128(
        (AS1 v4i*)(gsrc), (AS3 v4i*)(lds_dst), 0, 0);
#else
    *reinterpret_cast<float4*>(lds_dst) =
        *reinterpret_cast<const float4*>(gsrc);
#endif
}

#if HAVE_ASYNC_LDS
#  if HAVE_WAIT_BUILTIN
#    define WAIT_TILE() __builtin_amdgcn_s_wait_asynccnt(PER_TILE_ASYNC)
#    define WAIT_ALL()  __builtin_amdgcn_s_wait_asynccnt(0)
#  else
#    define WAIT_TILE() asm volatile("s_wait_asynccnt %0" :: "i"(PER_TILE_ASYNC) : "memory")
#    define WAIT_ALL()  asm volatile("s_wait_asynccnt %0" :: "i"(0) : "memory")
#  endif
#else
#  define WAIT_TILE() ((void)0)
#  define WAIT_ALL()  ((void)0)
#endif

// Issue one tile's worth of async global->LDS copies (both arrays).
__device__ __forceinline__ void issue_tile(const float* __restrict__ gp,
                                           const float* __restrict__ gm,
                                           float* lp, float* lm,
                                           int tile, int tid) {
    const char* gsp = (const char*)gp + (size_t)tile * TILE_BYTES;
    const char* gsm = (const char*)gm + (size_t)tile * TILE_BYTES;
    char* lpc = (char*)lp;
    char* lmc = (char*)lm;
    for (int i = tid * 16; i < TILE_BYTES; i += NTHR * 16)
        async_cp16(lpc + i, gsp + i);
    for (int i = tid * 16; i < TILE_BYTES; i += NTHR * 16)
        async_cp16(lmc + i, gsm + i);
}

// Per-cell YOLOv1 loss terms. acc: 0=pos_resp, 1=neg_resp, 2=pos_cls, 3=pos_off
__device__ __forceinline__ void cell_loss(const float* __restrict__ p,
                                          const float* __restrict__ t,
                                          float acc[4]) {
    const float tr0 = t[4], tr1 = t[9];
    const float pos = (tr0 + tr1 > 0.9f) ? 1.0f : 0.0f;

    // positive-cell class loss (channels 10..29)
    float cls = 0.0f;
#pragma unroll
    for (int c = 10; c < 30; ++c) {
        float d = p[c] - t[c];
        cls = fmaf(d, d, cls);
    }
    acc[2] += pos * cls;

    // negative response loss (per predictor)
    const float dr0 = p[4] - tr0;
    const float dr1 = p[9] - tr1;
    const float neg0 = (tr0 < 1.0f) ? 1.0f : 0.0f;
    const float neg1 = (tr1 < 1.0f) ? 1.0f : 0.0f;
    acc[1] += neg0 * dr0 * dr0 + neg1 * dr1 * dr1;

    // IoU(decode(t_box_j), decode(p_box_j))
    float iou[2];
#pragma unroll
    for (int j = 0; j < 2; ++j) {
        const int o = j * 5;
        const float thw = t[o + 2] * t[o + 2] * 0.5f;
        const float thh = t[o + 3] * t[o + 3] * 0.5f;
        const float ax0 = t[o + 0] - thw, ay0 = t[o + 1] - thh;
        const float ax1 = t[o + 0] + thw, ay1 = t[o + 1] + thh;
        const float phw = p[o + 2] * p[o + 2] * 0.5f;
        const float phh = p[o + 3] * p[o + 3] * 0.5f;
        const float bx0 = p[o + 0] - phw, by0 = p[o + 1] - phh;
        const float bx1 = p[o + 0] + phw, by1 = p[o + 1] + phh;
        const float x0 = fmaxf(ax0, bx0), y0 = fmaxf(ay0, by0);
        const float x1 = fminf(ax1, bx1), y1 = fminf(ay1, by1);
        const float inter = fmaxf(x1 - x0, 0.0f) * fmaxf(y1 - y0, 0.0f);
        const float areaA = (ax1 - ax0) * (ay1 - ay0);
        const float areaB = (bx1 - bx0) * (by1 - by0);
        iou[j] = inter / (areaA + areaB - inter + 1e-10f);
    }

    const int best = (iou[1] > iou[0]) ? 1 : 0;   // argmax, first-on-tie
    const int ob = best * 5;
    const float dbr = p[ob + 4] - iou[best];      // p_resp[best] - iou[best]
    acc[0] += pos * dbr * dbr;

    float off = 0.0f;
#pragma unroll
    for (int k = 0; k < 4; ++k) {
        float d = p[ob + k] - t[ob + k];
        off = fmaf(d, d, off);
    }
    acc[3] += pos * off;
}

__device__ __forceinline__ float wave_sum(float v) {
    v += __shfl_xor(v, 16, 32);
    v += __shfl_xor(v, 8, 32);
    v += __shfl_xor(v, 4, 32);
    v += __shfl_xor(v, 2, 32);
    v += __shfl_xor(v, 1, 32);
    return v;
}

__global__ void __launch_bounds__(NTHR)
yolo_loss_main(const float* __restrict__ pred, const float* __restrict__ meta,
               float* __restrict__ partials, int numTiles) {
    __shared__ float bufP[2][TILE_FLOATS];
    __shared__ float bufM[2][TILE_FLOATS];
    __shared__ float red[2][4];

    const int tid = threadIdx.x;
    const int G   = gridDim.x;

    float acc[4] = {0.0f, 0.0f, 0.0f, 0.0f};

    int t = blockIdx.x;
    if (t < numTiles)
        issue_tile(pred, meta, bufP[0], bufM[0], t, tid);

    int pb = 0;
    while (t < numTiles) {
        const int nt = t + G;
        if (nt < numTiles) {
            issue_tile(pred, meta, bufP[pb ^ 1], bufM[pb ^ 1], nt, tid);
            WAIT_TILE();          // current tile complete; next tile in flight
        } else {
            WAIT_ALL();           // drain
        }
        __syncthreads();          // all waves' async copies visible in LDS

        cell_loss(&bufP[pb][tid * CH],          &bufM[pb][tid * CH],          acc);
        cell_loss(&bufP[pb][(tid + NTHR) * CH], &bufM[pb][(tid + NTHR) * CH], acc);

        __syncthreads();          // done reading buf[pb] before it is reissued
        t = nt;
        pb ^= 1;
    }

    // block reduction: wave32 shuffle -> LDS -> lane 0
    const int wave = tid >> 5;
    const int lane = tid & 31;
#pragma unroll
    for (int c = 0; c < 4; ++c)
        acc[c] = wave_sum(acc[c]);
    if (lane == 0) {
#pragma unroll
        for (int c = 0; c < 4; ++c)
            red[wave][c] = acc[c];
    }
    __syncthreads();
    if (tid == 0) {
#pragma unroll
        for (int c = 0; c < 4; ++c)
            partials[(size_t)blockIdx.x * 4 + c] = red[0][c] + red[1][c];
    }
}

// Deterministic final reduction + scaling. One block, 4 waves (one / component).
__global__ void __launch_bounds__(128)
yolo_loss_final(const float* __restrict__ partials, int nBlocks,
                float* __restrict__ out, float invB) {
    __shared__ float comp[4];
    const int wave = threadIdx.x >> 5;
    const int lane = threadIdx.x & 31;
    if (wave < 4) {
        float s = 0.0f;
        for (int r = lane; r < nBlocks; r += 32)
            s += partials[(size_t)r * 4 + wave];
        s = wave_sum(s);
        if (lane == 0) comp[wave] = s;
    }
    __syncthreads();
    if (threadIdx.x == 0) {
        const float loss_pos_resp = comp[0] * invB * 1.0f;  // L_OBJ
        const float loss_neg_resp = comp[1] * invB * 0.5f;  // L_NOOBJ
        const float loss_pos_cls  = comp[2] * invB;
        const float loss_pos_off  = comp[3] * invB * 5.0f;  // L_COORD
        out[0] = loss_neg_resp + loss_pos_resp + loss_pos_off + loss_pos_cls;
        out[1] = loss_pos_resp;
        out[2] = loss_neg_resp;
        out[3] = loss_pos_cls;
        out[4] = loss_pos_off;
    }
}

extern "C" void kernel_launch(void* const* d_in, const int* in_sizes, int n_in,
                              void* d_out, int out_size, void* d_ws, size_t ws_size,
                              hipStream_t stream) {
    const float* pred = (const float*)d_in[0];
    const float* meta = (const float*)d_in[1];
    float* out      = (float*)d_out;
    float* partials = (float*)d_ws;

    const int ncells   = in_sizes[0] / CH;      // 802816
    const int numTiles = ncells / TILE;         // 6272 (exact for B=1024,H=W=28)

    int grid = 1568;                            // 4 tiles per block
    if (grid > numTiles) grid = (numTiles > 0) ? numTiles : 1;

    yolo_loss_main<<<dim3(grid), dim3(NTHR), 0, stream>>>(pred, meta, partials, numTiles);
    yolo_loss_final<<<dim3(1), dim3(128), 0, stream>>>(partials, grid, out, 1.0f / 1024.0f);
}